// WiouWbceLoss_17867063951463
// MI455X (gfx1250) — compile-verified
//
#include <hip/hip_runtime.h>

typedef float v8f __attribute__((ext_vector_type(8)));
typedef float v2f __attribute__((ext_vector_type(2)));

#define HW      1024
#define TILE    64
#define HALO    15
#define HROWS   94          // TILE + 2*HALO
#define LSTRIDE 80          // padded LDS row stride (floats) -> conflict-free WMMA B loads
#define NWG     (32 * 256)  // 32 images * 16x16 tiles

// ---------------------------------------------------------------------------
// Per-tile kernel: fused 31x31 box filter (separable: scalar horizontal pass,
// WMMA vertical pass) + weighted BCE/IoU partial reductions.
// ---------------------------------------------------------------------------
__global__ __launch_bounds__(128)
void wiou_tile_kernel(const float* __restrict__ inp,
                      const float* __restrict__ tgt,
                      float* __restrict__ ws)
{
    __shared__ float sh[96 * LSTRIDE];   // horizontal sliding sums (96 rows: 94 data + 2 zero)
    __shared__ float red[4][3];

    const int wg   = blockIdx.x;
    const int b    = wg >> 8;
    const int tile = wg & 255;
    const int r0   = (tile >> 4) * TILE;
    const int c0   = (tile & 15) * TILE;
    const size_t base = (size_t)b * (size_t)(HW * HW);
    const float* T = tgt + base;
    const float* X = inp + base;

    const int t = threadIdx.x;

    // ---- prefetch the output-tile input/target lines (consumed in epilogue) ----
    {
        const int prow = r0 + (t >> 1);          // 64 rows, 2 threads per row
        const int pcol = c0 + (t & 1) * 32;      // 2 x 128B halves of the 256B row segment
        const size_t pidx = (size_t)prow * HW + pcol;
        __builtin_prefetch(X + pidx, 0, 0);      // -> global_prefetch_b8
        __builtin_prefetch(T + pidx, 0, 0);
    }

    // ---- horizontal 31-tap sliding sums into LDS (zero padding outside image) ----
    if (t < HROWS) {
        const int r = r0 - HALO + t;
        float* dst = &sh[t * LSTRIDE];
        if (r < 0 || r >= HW) {
            for (int i = 0; i < TILE; ++i) dst[i] = 0.0f;
        } else {
            const float* row = T + (size_t)r * HW;
            float s = 0.0f;
            #pragma unroll
            for (int k = -HALO; k <= HALO; ++k) {
                int c = c0 + k;
                s += (c >= 0 && c < HW) ? row[c] : 0.0f;
            }
            dst[0] = s;
            for (int i = 1; i < TILE; ++i) {
                int cadd = c0 + i + HALO;
                int csub = c0 + i - HALO - 1;
                float va = (cadd < HW)  ? row[cadd] : 0.0f;
                float vs = (csub >= 0)  ? row[csub] : 0.0f;
                s += va - vs;
                dst[i] = s;
            }
        }
    } else if (t < HROWS + 2) {
        // zero the two padded K rows so WMMA 0*x lanes see finite data
        float* dst = &sh[(HROWS + (t - HROWS)) * LSTRIDE];
        for (int i = 0; i < TILE; ++i) dst[i] = 0.0f;
    }
    __syncthreads();

    // ---- vertical 31-row sum via V_WMMA_F32_16X16X4_F32: D = Band(16x48) * Hsum(48x16) ----
    const int wave = t >> 5;          // 4 waves -> 4 sub-tile rows
    const int lane = t & 31;
    const int m    = lane & 15;       // A-matrix M row / B,N column index
    const int hh   = lane >> 4;       // half-wave select (K split per ISA layout)

    float bce_acc = 0.0f, inter_acc = 0.0f, union_acc = 0.0f;

    const int sm = wave;              // sub-tile row 0..3
    for (int sn = 0; sn < 4; ++sn) {
        v8f acc = {};
        const int col = sn * 16 + m;
        #pragma unroll
        for (int q = 0; q < 12; ++q) {          // K = 48 in chunks of 4
            const int k0 = 4 * q + 2 * hh;      // A layout: lanes16-31 hold K=2,3
            const int k1 = k0 + 1;
            v2f a;
            a.x = (k0 >= m && k0 <= m + 30) ? 1.0f : 0.0f;   // banded ones matrix
            a.y = (k1 >= m && k1 <= m + 30) ? 1.0f : 0.0f;
            const int rowb = sm * 16 + k0;      // B layout: row K, column N per lane
            v2f bb;
            bb.x = sh[rowb * LSTRIDE + col];
            bb.y = sh[(rowb + 1) * LSTRIDE + col];
            acc = __builtin_amdgcn_wmma_f32_16x16x4_f32(
                false, a, false, bb, (short)0, acc, false, false);
        }

        // ---- epilogue: D layout -> pixels; fuse weit/BCE/IoU ----
        const int gcol = c0 + col;
        #pragma unroll
        for (int v = 0; v < 8; ++v) {
            const int grow = r0 + sm * 16 + v + 8 * hh;   // VGPR v: M=v (lanes0-15), v+8 (16-31)
            const size_t idx = (size_t)grow * HW + gcol;
            const float tv  = T[idx];
            const float xv  = X[idx];
            const float box = acc[v] * (1.0f / 961.0f);
            const float weit = 1.0f + 5.0f * fabsf(box - tv);
            const float e   = __expf(-fabsf(xv));
            const float inv = __builtin_amdgcn_rcpf(1.0f + e);   // v_rcp_f32, no div sequence
            bce_acc  += fmaxf(xv, 0.0f) - xv * tv + __logf(1.0f + e);
            const float p = (xv >= 0.0f) ? inv : e * inv;        // sigmoid, stable
            inter_acc += p * tv * weit;
            union_acc += (p + tv) * weit;
        }
    }

    // ---- fixed-order reduction (no FP atomics -> deterministic) ----
    #pragma unroll
    for (int off = 16; off > 0; off >>= 1) {
        bce_acc   += __shfl_down(bce_acc,   off, 32);
        inter_acc += __shfl_down(inter_acc, off, 32);
        union_acc += __shfl_down(union_acc, off, 32);
    }
    if (lane == 0) {
        red[wave][0] = bce_acc;
        red[wave][1] = inter_acc;
        red[wave][2] = union_acc;
    }
    __syncthreads();
    if (t == 0) {
        float s0 = 0.0f, s1 = 0.0f, s2 = 0.0f;
        for (int w = 0; w < 4; ++w) { s0 += red[w][0]; s1 += red[w][1]; s2 += red[w][2]; }
        ws[wg]            = s0;   // bce partial
        ws[NWG + wg]      = s1;   // inter partial
        ws[2 * NWG + wg]  = s2;   // union partial
    }
}

// ---------------------------------------------------------------------------
// Final reduction: scalar = mean_b( bce + wiou_b )   (wbce == bce exactly)
// ---------------------------------------------------------------------------
__global__ __launch_bounds__(256)
void wiou_finalize_kernel(const float* __restrict__ ws, float* __restrict__ out)
{
    __shared__ float sb[256];
    __shared__ float swiou[32];
    const int t = threadIdx.x;

    float s = 0.0f;
    for (int i = t; i < NWG; i += 256) s += ws[i];
    sb[t] = s;
    __syncthreads();
    #pragma unroll
    for (int off = 128; off > 0; off >>= 1) {
        if (t < off) sb[t] += sb[t + off];
        __syncthreads();
    }

    if (t < 32) {
        float inter = 0.0f, uni = 0.0f;
        for (int i = 0; i < 256; ++i) {
            inter += ws[NWG     + t * 256 + i];
            uni   += ws[2 * NWG + t * 256 + i];
        }
        swiou[t] = 1.0f - (inter + 1.0f) / (uni - inter + 1.0f);
    }
    __syncthreads();

    if (t == 0) {
        const float bce = sb[0] / (float)((size_t)32 * HW * HW);
        float w = 0.0f;
        for (int i = 0; i < 32; ++i) w += swiou[i];
        out[0] = bce + w * (1.0f / 32.0f);
    }
}

extern "C" void kernel_launch(void* const* d_in, const int* in_sizes, int n_in,
                              void* d_out, int out_size, void* d_ws, size_t ws_size,
                              hipStream_t stream) {
    const float* inp = (const float*)d_in[0];   // input (logits)
    const float* tgt = (const float*)d_in[1];   // target
    float* out = (float*)d_out;                  // 1 float
    float* ws  = (float*)d_ws;                   // needs 3*8192*4 = 96 KB

    wiou_tile_kernel<<<NWG, 128, 0, stream>>>(inp, tgt, ws);
    wiou_finalize_kernel<<<1, 256, 0, stream>>>(ws, out);
}